// DSGCN_51213190037829
// MI455X (gfx1250) — compile-verified
//
#include <hip/hip_runtime.h>
#include <stdint.h>

// Problem constants (fixed by the reference)
#define B_ 8
#define N_ 4096
#define D_ 768

typedef __attribute__((ext_vector_type(2))) float v2f;
typedef __attribute__((ext_vector_type(8))) float v8f;

// ---- CDNA5 async global->LDS copy (ASYNCcnt tracked) --------------------
__device__ __forceinline__ void async_ld_b128(uint32_t lds_byte_off, const float* gptr) {
    // LDS byte address comes from the first VGPR operand; the global address is 64-bit
    asm volatile("global_load_async_to_lds_b128 %0, %1, off"
                 :: "v"(lds_byte_off), "v"(gptr)
                 : "memory");
}
__device__ __forceinline__ void wait_async() {
    asm volatile("s_wait_asynccnt 0" ::: "memory");
}

// ---- row-degree kernel: inv[b*N_+n] = 1/(sum_m adj + 1) -----------------
__global__ __launch_bounds__(256) void rowsum_kernel(const float* __restrict__ adj,
                                                     float* __restrict__ invd) {
    const int row = blockIdx.x;                 // 0 .. B_*N_-1
    const float4* rp = (const float4*)(adj + (size_t)row * N_);
    float s = 0.f;
    for (int i = threadIdx.x; i < N_ / 4; i += 256) {
        float4 v = rp[i];
        s += v.x + v.y + v.z + v.w;
    }
    __shared__ float red[256];
    red[threadIdx.x] = s;
    __syncthreads();
    for (int off = 128; off > 0; off >>= 1) {
        if (threadIdx.x < off) red[threadIdx.x] += red[threadIdx.x + off];
        __syncthreads();
    }
    if (threadIdx.x == 0) invd[row] = 1.0f / (red[0] + 1.0f);
}

// ---- tiled WMMA f32 GEMM ------------------------------------------------
// C[M x 768] = A[M x K] * B + epilogue
//   BT = true : B given as [Ncols x K] row-major (i.e. we need B^T; W0 / Wout)
//   BT = false: B given as [K x 768] row-major (xW)
// EP = 0: C = A*B + bias          (stage 1, C = xW -> d_out)
// EP = 1: C = relu((A*B + xW)*inv) + nodes  (stage 2, batched over blockIdx.z)
// EP = 2: C = A*B + bias          (stage 3, C = out -> d_out)
template <int EP, bool BT>
__global__ __launch_bounds__(256) void gemm_kernel(const float* __restrict__ A, int lda,
                                                   const float* __restrict__ Bm, int ldb,
                                                   float* __restrict__ C, int K,
                                                   const float* __restrict__ bias,
                                                   const float* __restrict__ nodesp,
                                                   const float* __restrict__ invp) {
    __shared__ __align__(16) float smA[64 * 32];    //  8 KB
    __shared__ __align__(16) float smB[32 * 128];   // 16 KB (either layout)
    __shared__ float smInv[64];

    const int tid  = threadIdx.x;
    const int lane = tid & 31;
    const int wave = tid >> 5;
    const int half = lane >> 4;   // 0: lanes 0-15, 1: lanes 16-31
    const int lm   = lane & 15;
    const int wm   = wave & 3;    // 4 row groups of 16 -> BM = 64
    const int wn   = wave >> 2;   // 2 col groups of 64 -> BN = 128
    const int rowBase = wm * 16;
    const int colW    = wn * 64;

    const int blockRow = blockIdx.x;
    const int blockCol = blockIdx.y;

    if (EP == 1) {
        const size_t bz = blockIdx.z;
        A      += bz * ((size_t)N_ * N_);
        Bm     += bz * ((size_t)N_ * D_);
        C      += bz * ((size_t)N_ * D_);
        nodesp += bz * ((size_t)N_ * D_);
        invp   += bz * (size_t)N_;
        if (tid < 64) smInv[tid] = invp[blockRow * 64 + tid];
    }

    const uint32_t smA_off = (uint32_t)(uintptr_t)(&smA[0]);
    const uint32_t smB_off = (uint32_t)(uintptr_t)(&smB[0]);

    v8f acc[4];
    const v8f vzero = {0.f, 0.f, 0.f, 0.f, 0.f, 0.f, 0.f, 0.f};
#pragma unroll
    for (int c = 0; c < 4; ++c) acc[c] = vzero;

    for (int k0 = 0; k0 < K; k0 += 32) {
        // ---- stage A tile 64x32 (512 x 16B chunks, 2 per thread) ----
#pragma unroll
        for (int i = 0; i < 2; ++i) {
            int idx = tid + i * 256;
            int r   = idx >> 3;
            int c4  = (idx & 7) << 2;
            const float* gp = A + (size_t)(blockRow * 64 + r) * lda + (k0 + c4);
            async_ld_b128(smA_off + (uint32_t)((r * 32 + c4) * 4), gp);
        }
        // ---- stage B tile 32x128 (1024 x 16B chunks, 4 per thread) ----
#pragma unroll
        for (int i = 0; i < 4; ++i) {
            int idx = tid + i * 256;
            if (BT) {
                int n  = idx >> 3;            // 0..127 output column within tile
                int c4 = (idx & 7) << 2;      // k chunk
                const float* gp = Bm + (size_t)(blockCol * 128 + n) * ldb + (k0 + c4);
                async_ld_b128(smB_off + (uint32_t)((n * 32 + c4) * 4), gp);  // smB[n][k]
            } else {
                int kr = idx >> 5;            // 0..31 k row
                int c4 = (idx & 31) << 2;     // col chunk
                const float* gp = Bm + (size_t)(k0 + kr) * ldb + (blockCol * 128 + c4);
                async_ld_b128(smB_off + (uint32_t)((kr * 128 + c4) * 4), gp); // smB[k][n]
            }
        }
        wait_async();
        __syncthreads();

        // ---- compute: 8 k-steps x 4 column tiles of v_wmma_f32_16x16x4_f32 ----
#pragma unroll
        for (int kk = 0; kk < 32; kk += 4) {
            // A frag 16x4: lanes 0-15 -> (M=lm, K=kk,kk+1); lanes 16-31 -> K=kk+2,kk+3
            v2f a = *(const v2f*)&smA[(rowBase + lm) * 32 + kk + 2 * half];
#pragma unroll
            for (int c = 0; c < 4; ++c) {
                v2f bf;
                if (BT) {
                    bf = *(const v2f*)&smB[(colW + c * 16 + lm) * 32 + kk + 2 * half];
                } else {
                    bf.x = smB[(kk + 2 * half) * 128 + colW + c * 16 + lm];
                    bf.y = smB[(kk + 2 * half + 1) * 128 + colW + c * 16 + lm];
                }
                acc[c] = __builtin_amdgcn_wmma_f32_16x16x4_f32(
                    false, a, false, bf, (short)0, acc[c], false, false);
            }
        }
        __syncthreads();
    }

    // ---- epilogue ----
#pragma unroll
    for (int c = 0; c < 4; ++c) {
        const int colG = blockCol * 128 + colW + c * 16 + lm;
        float bb = 0.f;
        if (EP != 1) bb = bias[colG];
#pragma unroll
        for (int r = 0; r < 8; ++r) {
            const int gr = blockRow * 64 + rowBase + half * 8 + r;  // (batch-local) row
            const float v = acc[c][r];
            if (EP == 1) {
                const float xw  = Bm[(size_t)gr * D_ + colG];       // Bm == xW (batch-adjusted)
                const float nd  = nodesp[(size_t)gr * D_ + colG];
                const float inv = smInv[rowBase + half * 8 + r];
                float t = (v + xw) * inv;
                t = t > 0.f ? t : 0.f;
                C[(size_t)gr * D_ + colG] = t + nd;
            } else {
                C[(size_t)gr * D_ + colG] = v + bb;
            }
        }
    }
}

extern "C" void kernel_launch(void* const* d_in, const int* in_sizes, int n_in,
                              void* d_out, int out_size, void* d_ws, size_t ws_size,
                              hipStream_t stream) {
    (void)in_sizes; (void)n_in; (void)out_size; (void)ws_size;
    const float* nodes = (const float*)d_in[0];
    const float* adj   = (const float*)d_in[1];
    const float* W0    = (const float*)d_in[2];
    const float* b0    = (const float*)d_in[3];
    const float* Wout  = (const float*)d_in[4];
    const float* bout  = (const float*)d_in[5];
    float* out  = (float*)d_out;
    float* invd = (float*)d_ws;                 // B_*N_ floats (128 KB)
    float* gcn  = (float*)d_ws + (size_t)B_ * N_;  // B_*N_*D_ floats (~100.7 MB)

    // 1) row degrees
    rowsum_kernel<<<dim3(B_ * N_), 256, 0, stream>>>(adj, invd);

    // 2) xW = nodes @ W0^T + b0  -> d_out (scratch; exactly out-sized)
    gemm_kernel<0, true><<<dim3((B_ * N_) / 64, D_ / 128, 1), 256, 0, stream>>>(
        nodes, D_, W0, D_, out, D_, b0, nullptr, nullptr);

    // 3) gcn = relu((adj @ xW + xW) * inv) + nodes  -> ws
    gemm_kernel<1, false><<<dim3(N_ / 64, D_ / 128, B_), 256, 0, stream>>>(
        adj, N_, out, D_, gcn, N_, nullptr, nodes, invd);

    // 4) out = gcn @ Wout^T + bout -> d_out
    gemm_kernel<2, true><<<dim3((B_ * N_) / 64, D_ / 128, 1), 256, 0, stream>>>(
        gcn, D_, Wout, D_, out, D_, bout, nullptr, nullptr);
}